// CountingDiceLoss_47201690583818
// MI455X (gfx1250) — compile-verified
//
#include <hip/hip_runtime.h>
#include <hip/hip_bf16.h>
#include <stdint.h>

typedef __attribute__((ext_vector_type(2))) float v2f;
typedef __attribute__((ext_vector_type(8))) float v8f;

#define BB 8
#define HH 512
#define WW 512
#define PP 40

// acc layout per batch (8 floats, 6 used): [s_py, s_p, s_y, s_sq, s_x2, s_dm]
__global__ void cdl_zero_acc(float* __restrict__ acc) {
    if (threadIdx.x < 64) acc[threadIdx.x] = 0.0f;
}

// Build separable Gaussian factor matrices:
//   Amat[b][i][p] = exp(-0.5*(i-ci)^2)                      (row factor)
//   Bt  [b][j][p] = valid ? exp(-0.5*(j-cj)^2)*invnorm : 0  (col factor, transposed, norm folded)
__global__ void cdl_prep_AB(const int* __restrict__ cent, const uint8_t* __restrict__ valid,
                            float* __restrict__ Amat, float* __restrict__ Bt) {
    int t = blockIdx.x * blockDim.x + threadIdx.x;
    if (t >= 2 * BB * HH) return;
    int half = t / (BB * HH);
    int r    = t % (BB * HH);
    int b    = r / HH;
    int idx  = r % HH;
    // reference: g = exp(-d2/2)/2.5066283, then /2.50635  (s_k == 1.0f in f32)
    const float invnorm = 1.0f / (2.50662827f * 2.50635f);
    if (half == 0) {
        float* dst = Amat + (size_t)r * PP;
        #pragma unroll 4
        for (int p = 0; p < PP; ++p) {
            float d = (float)idx - (float)cent[(b * PP + p) * 2 + 0];
            dst[p] = __expf(-0.5f * d * d);
        }
    } else {
        float* dst = Bt + (size_t)r * PP;
        #pragma unroll 4
        for (int p = 0; p < PP; ++p) {
            float d = (float)idx - (float)cent[(b * PP + p) * 2 + 1];
            float e = __expf(-0.5f * d * d) * invnorm;
            dst[p] = valid[b * PP + p] ? e : 0.0f;
        }
    }
}

// Fused: each wave computes a 16x16 density-map tile via V_WMMA_F32_16X16X4_F32
// over K=40 (10 steps), then consumes x/y/mask for those pixels and reduces.
// Block = 128 threads = 4 waves -> 16 rows x 64 cols per block.
__global__ void __launch_bounds__(128)
cdl_fused_main(const float* __restrict__ x, const float* __restrict__ y,
               const float* __restrict__ mask, const float* __restrict__ Amat,
               const float* __restrict__ Bt, float* __restrict__ acc) {
    const int b    = blockIdx.z;
    const int wave = threadIdx.x >> 5;
    const int lane = threadIdx.x & 31;
    const int row0 = blockIdx.y * 16;
    const int col0 = blockIdx.x * 64 + wave * 16;
    const int mn   = lane & 15;   // M for A-frag, N for B/C-frags
    const int h    = lane >> 4;   // lane-half selects K pair (A/B) and M+8 (C/D)

    const float* Arow = Amat + ((size_t)b * HH + (row0 + mn)) * PP;
    const float* Brow = Bt   + ((size_t)b * WW + (col0 + mn)) * PP;

    v8f c = {};
    #pragma unroll
    for (int ks = 0; ks < PP / 4; ++ks) {
        const int k0 = ks * 4 + 2 * h;
        v2f a, bb;
        a.x  = Arow[k0];  a.y  = Arow[k0 + 1];
        bb.x = Brow[k0];  bb.y = Brow[k0 + 1];
        // D = A(16x4) * B(4x16) + C, f32
        c = __builtin_amdgcn_wmma_f32_16x16x4_f32(false, a, false, bb,
                                                  (short)0, c, false, false);
    }

    // Consume the 16x16 tile: lane handles (i = row0 + v + 8h, j = col0 + mn), v=0..7
    float s_py = 0.f, s_p = 0.f, s_y = 0.f, s_sq = 0.f, s_x2 = 0.f, s_dm = 0.f;
    const size_t xb = (size_t)b * 3 * HH * WW;
    const size_t yb = (size_t)b * HH * WW;
    const size_t pl = (size_t)HH * WW;
    #pragma unroll
    for (int v = 0; v < 8; ++v) {
        const int    i   = row0 + v + 8 * h;
        const int    j   = col0 + mn;
        const size_t pix = (size_t)i * WW + j;
        const float x0 = x[xb + pix];
        const float x1 = x[xb + pl + pix];
        const float x2 = x[xb + 2 * pl + pix];
        const float yv = y[yb + pix];
        const float mk = mask[yb + pix];
        const float dm = c[v] * mk;                       // masked density map
        const float p1 = 1.0f / (1.0f + __expf(x0 - x1)); // softmax ch1 of {x0,x1}
        s_py += p1 * yv;
        s_p  += p1;
        s_y  += yv;
        const float d = x2 - dm;
        s_sq += d * d;
        s_x2 += x2;
        s_dm += dm;
    }

    // wave32 butterfly reduction
    #pragma unroll
    for (int off = 16; off > 0; off >>= 1) {
        s_py += __shfl_xor(s_py, off, 32);
        s_p  += __shfl_xor(s_p,  off, 32);
        s_y  += __shfl_xor(s_y,  off, 32);
        s_sq += __shfl_xor(s_sq, off, 32);
        s_x2 += __shfl_xor(s_x2, off, 32);
        s_dm += __shfl_xor(s_dm, off, 32);
    }
    if (lane == 0) {
        float* ab = acc + b * 8;
        atomicAdd(ab + 0, s_py);
        atomicAdd(ab + 1, s_p);
        atomicAdd(ab + 2, s_y);
        atomicAdd(ab + 3, s_sq);
        atomicAdd(ab + 4, s_x2);
        atomicAdd(ab + 5, s_dm);
    }
}

__global__ void cdl_finalize(const float* __restrict__ acc, float* __restrict__ out) {
    if (threadIdx.x == 0 && blockIdx.x == 0) {
        float dsum = 0.f, sq = 0.f, sx2 = 0.f, sdm = 0.f;
        #pragma unroll
        for (int b = 0; b < BB; ++b) {
            const float* a = acc + b * 8;
            dsum += (2.0f * a[0] + 1e-5f) / (a[1] + a[2] + 1e-5f);
            sq  += a[3];
            sx2 += a[4];
            sdm += a[5];
        }
        const float l_dice = -dsum / (float)BB;
        const float l_dm   = sq / ((float)BB * (float)HH * (float)WW);
        const float dn     = sx2 - sdm;
        out[0] = l_dice + l_dm + dn * dn;
    }
}

extern "C" void kernel_launch(void* const* d_in, const int* in_sizes, int n_in,
                              void* d_out, int out_size, void* d_ws, size_t ws_size,
                              hipStream_t stream) {
    const float*   x     = (const float*)d_in[0];
    const float*   y     = (const float*)d_in[1];
    const float*   mask  = (const float*)d_in[2];
    const int*     cent  = (const int*)d_in[3];
    const uint8_t* valid = (const uint8_t*)d_in[4];

    float* acc  = (float*)d_ws;                       // 64 floats
    float* Amat = acc + 64;                           // [B][H][P]
    float* Bt   = Amat + (size_t)BB * HH * PP;        // [B][W][P]

    cdl_zero_acc<<<1, 64, 0, stream>>>(acc);
    cdl_prep_AB<<<(2 * BB * HH + 255) / 256, 256, 0, stream>>>(cent, valid, Amat, Bt);
    dim3 grid(WW / 64, HH / 16, BB);
    cdl_fused_main<<<grid, 128, 0, stream>>>(x, y, mask, Amat, Bt, acc);
    cdl_finalize<<<1, 32, 0, stream>>>(acc, (float*)d_out);
}